// GCN_48155173322928
// MI455X (gfx1250) — compile-verified
//
#include <hip/hip_runtime.h>
#include <hip/hip_bf16.h>
#include <math.h>

#define N_NODES 100000
#define N_EDGES 1600000
#define N_FEAT  512
#define N_HID   64
#define N_CLASS 40

typedef __attribute__((ext_vector_type(2))) float v2f;
typedef __attribute__((ext_vector_type(8))) float v8f;

// ---------------------------------------------------------------------------
// out[i] = b[i % D]   (pre-broadcast bias into the SpMM accumulator buffers)
// ---------------------------------------------------------------------------
__global__ void init_bias_kernel(float* __restrict__ out, const float* __restrict__ b,
                                 int D, long long total) {
    long long i = (long long)blockIdx.x * blockDim.x + threadIdx.x;
    long long stride = (long long)gridDim.x * blockDim.x;
    for (; i < total; i += stride) out[i] = b[(int)(i % D)];
}

// ---------------------------------------------------------------------------
// H0 = X @ W1   (100000x512 @ 512x64), fp32 WMMA 16x16x4.
// Block = 256 threads = 8 waves; each wave owns a 16-row tile and all 64 cols
// (4 accumulators). W1 staged into LDS in 128-row K-chunks (32 KB),
// PAIR-INTERLEAVED: sB[(k/2)*64 + col] = {W1[k][col], W1[k+1][col]} so each
// B fragment is a single aligned ds_load_b64 (no repack movs before WMMA).
// ---------------------------------------------------------------------------
__global__ __launch_bounds__(256) void gemm1_kernel(const float* __restrict__ X,
                                                    const float* __restrict__ W1,
                                                    float* __restrict__ H0) {
    __shared__ v2f sB[64 * N_HID];  // 32 KB K-chunk of W1, pair-interleaved

    const int wv   = threadIdx.x >> 5;
    const int lane = threadIdx.x & 31;
    const int tileM = blockIdx.x * 128 + wv * 16;
    const bool active = (tileM < N_NODES);   // wave-uniform (EXEC all-ones in WMMA region)

    const int mrow = lane & 15;              // A-frag: M = lane&15
    const int koff = (lane >> 4) << 1;       // A-frag: K = 2*(lane>>4) .. +1
    const int col  = lane & 15;              // B/D-frag: N = lane&15
    const float* pA = X + (size_t)(tileM + mrow) * N_FEAT;

    v8f c0 = {}, c1 = {}, c2 = {}, c3 = {};

    for (int kc = 0; kc < N_FEAT; kc += 128) {
        __syncthreads();
        // cooperative coalesced stage: each thread reads 8 float4 of W1 and
        // scatters pair-interleaved into LDS.
        {
            const float4* g4 = (const float4*)(W1 + (size_t)kc * N_HID);
            float* sBf = (float*)sB;
#pragma unroll
            for (int i = 0; i < 8; ++i) {
                const int idx = threadIdx.x + 256 * i;   // float4 index in chunk
                const int k   = idx >> 4;                // row 0..127
                const int c0i = (idx & 15) << 2;         // col 0..60
                float4 vdat = g4[idx];
                const int base = ((k >> 1) * N_HID + c0i) * 2 + (k & 1);
                sBf[base + 0] = vdat.x;
                sBf[base + 2] = vdat.y;
                sBf[base + 4] = vdat.z;
                sBf[base + 6] = vdat.w;
            }
        }
        __syncthreads();

        if (active) {
            // overlap next K-chunk of this wave's A rows with current compute
            if (kc + 128 < N_FEAT) __builtin_prefetch(pA + kc + 128 + koff, 0, 0);
#pragma unroll 8
            for (int k = 0; k < 128; k += 4) {
                v2f a = *(const v2f*)(pA + kc + k + koff);         // 8B-aligned
                const v2f* bp = sB + ((k + koff) >> 1) * N_HID;    // k+koff even
                v2f b;
                b = bp[col];
                c0 = __builtin_amdgcn_wmma_f32_16x16x4_f32(false, a, false, b, (short)0, c0, false, false);
                b = bp[16 + col];
                c1 = __builtin_amdgcn_wmma_f32_16x16x4_f32(false, a, false, b, (short)0, c1, false, false);
                b = bp[32 + col];
                c2 = __builtin_amdgcn_wmma_f32_16x16x4_f32(false, a, false, b, (short)0, c2, false, false);
                b = bp[48 + col];
                c3 = __builtin_amdgcn_wmma_f32_16x16x4_f32(false, a, false, b, (short)0, c3, false, false);
            }
        }
    }

    if (active) {
        // D layout: VGPR v -> row tileM + v + 8*(lane>>4), col = lane&15
        const int rbase = tileM + ((lane >> 4) << 3);
#pragma unroll
        for (int v = 0; v < 8; ++v) {
            float* o = H0 + (size_t)(rbase + v) * N_HID + col;
            o[0]  = c0[v];
            o[16] = c1[v];
            o[32] = c2[v];
            o[48] = c3[v];
        }
    }
}

// ---------------------------------------------------------------------------
// Z = relu(H) @ W2   (100000x64 @ 64x40), N padded 40->48 (3 WMMA tiles),
// ReLU fused into the A-fragment load; W2 staged pair-interleaved in LDS;
// masked store of last tile.
// ---------------------------------------------------------------------------
__global__ __launch_bounds__(256) void gemm2_kernel(const float* __restrict__ H,
                                                    const float* __restrict__ W2,
                                                    float* __restrict__ Z) {
    __shared__ v2f sB[32 * 48];  // 12 KB: K-pairs (32) x padded cols (48)

    for (int i = threadIdx.x; i < 32 * 48; i += 256) {
        const int kp = i / 48, cc = i - kp * 48;
        v2f w;
        if (cc < N_CLASS) {
            w.x = W2[(2 * kp) * N_CLASS + cc];
            w.y = W2[(2 * kp + 1) * N_CLASS + cc];
        } else {
            w.x = 0.0f; w.y = 0.0f;
        }
        sB[kp * 48 + cc] = w;
    }
    __syncthreads();

    const int wv   = threadIdx.x >> 5;
    const int lane = threadIdx.x & 31;
    const int tileM = blockIdx.x * 128 + wv * 16;
    if (tileM >= N_NODES) return;     // no barriers after this point

    const int mrow = lane & 15;
    const int koff = (lane >> 4) << 1;
    const int col  = lane & 15;
    const float* pA = H + (size_t)(tileM + mrow) * N_HID;

    v8f c0 = {}, c1 = {}, c2 = {};
#pragma unroll 4
    for (int k = 0; k < N_HID; k += 4) {
        v2f a = *(const v2f*)(pA + k + koff);   // 8B-aligned b64 load
        a.x = fmaxf(a.x, 0.0f);                 // fused ReLU
        a.y = fmaxf(a.y, 0.0f);
        const v2f* bp = sB + ((k + koff) >> 1) * 48;
        v2f b;
        b = bp[col];
        c0 = __builtin_amdgcn_wmma_f32_16x16x4_f32(false, a, false, b, (short)0, c0, false, false);
        b = bp[16 + col];
        c1 = __builtin_amdgcn_wmma_f32_16x16x4_f32(false, a, false, b, (short)0, c1, false, false);
        b = bp[32 + col];
        c2 = __builtin_amdgcn_wmma_f32_16x16x4_f32(false, a, false, b, (short)0, c2, false, false);
    }

    const int rbase = tileM + ((lane >> 4) << 3);
#pragma unroll
    for (int v = 0; v < 8; ++v) {
        float* o = Z + (size_t)(rbase + v) * N_CLASS;
        o[col]      = c0[v];
        o[16 + col] = c1[v];
        if (col < 8) o[32 + col] = c2[v];      // cols 32..39 valid only
    }
}

// ---------------------------------------------------------------------------
// out[dst] += val * in[src]  with dst sorted: register run-accumulation,
// atomic flush only on run boundaries. Block = 64 threads (thread = feature),
// 512 edges per block, edge triples staged through LDS.
// ---------------------------------------------------------------------------
template <int D>
__global__ __launch_bounds__(64) void spmm_kernel(const float* __restrict__ vals,
                                                  const int* __restrict__ src,
                                                  const int* __restrict__ dst,
                                                  const float* __restrict__ in,
                                                  float* __restrict__ out) {
    constexpr int EPB = 512, CH = 256;
    __shared__ int   s_src[CH];
    __shared__ int   s_dst[CH];
    __shared__ float s_val[CH];

    const int f = threadIdx.x;
    const int base0 = blockIdx.x * EPB;
    const int end = min(base0 + EPB, N_EDGES);

    float acc = 0.0f;
    int cur = -1;

    for (int base = base0; base < end; base += CH) {
        const int cnt = min(CH, end - base);
        __syncthreads();
        for (int i = threadIdx.x; i < cnt; i += 64) {
            s_src[i] = src[base + i];
            s_dst[i] = dst[base + i];
            s_val[i] = vals[base + i];
        }
        __syncthreads();
        if (f < D) {
            for (int i = 0; i < cnt; ++i) {
                const int d = s_dst[i];
                if (d != cur) {
                    if (cur >= 0) atomicAdd(out + (size_t)cur * D + f, acc);
                    cur = d;
                    acc = 0.0f;
                }
                acc = fmaf(s_val[i], in[(size_t)s_src[i] * D + f], acc);
            }
        }
    }
    if (f < D && cur >= 0) atomicAdd(out + (size_t)cur * D + f, acc);
}

// ---------------------------------------------------------------------------
// In-place log_softmax over 40 classes, thread per node.
// ---------------------------------------------------------------------------
__global__ void logsoftmax_kernel(float* __restrict__ out) {
    const int n = blockIdx.x * blockDim.x + threadIdx.x;
    if (n >= N_NODES) return;
    float* row = out + (size_t)n * N_CLASS;
    float v[N_CLASS];
    float m = -INFINITY;
#pragma unroll
    for (int c = 0; c < N_CLASS; ++c) { v[c] = row[c]; m = fmaxf(m, v[c]); }
    float s = 0.0f;
#pragma unroll
    for (int c = 0; c < N_CLASS; ++c) s += expf(v[c] - m);
    const float lse = m + logf(s);
#pragma unroll
    for (int c = 0; c < N_CLASS; ++c) row[c] = v[c] - lse;
}

// ---------------------------------------------------------------------------
extern "C" void kernel_launch(void* const* d_in, const int* in_sizes, int n_in,
                              void* d_out, int out_size, void* d_ws, size_t ws_size,
                              hipStream_t stream) {
    const float* x   = (const float*)d_in[0];
    const float* W1  = (const float*)d_in[1];
    const float* b1  = (const float*)d_in[2];
    const float* W2  = (const float*)d_in[3];
    const float* b2  = (const float*)d_in[4];
    const float* av  = (const float*)d_in[5];
    const int*   src = (const int*)d_in[6];
    const int*   dst = (const int*)d_in[7];
    float* out = (float*)d_out;

    float* h0 = (float*)d_ws;                         // [N, 64]  25.6 MB
    float* h  = h0 + (size_t)N_NODES * N_HID;         // [N, 64]  25.6 MB
    float* z  = h  + (size_t)N_NODES * N_HID;         // [N, 40]  16.0 MB

    // accumulator buffers pre-seeded with biases
    init_bias_kernel<<<512, 256, 0, stream>>>(h, b1, N_HID, (long long)N_NODES * N_HID);
    init_bias_kernel<<<512, 256, 0, stream>>>(out, b2, N_CLASS, (long long)N_NODES * N_CLASS);

    // h0 = X @ W1
    gemm1_kernel<<<(N_NODES + 127) / 128, 256, 0, stream>>>(x, W1, h0);
    // h (= b1) += A @ h0
    spmm_kernel<N_HID><<<(N_EDGES + 511) / 512, 64, 0, stream>>>(av, src, dst, h0, h);
    // z = relu(h) @ W2
    gemm2_kernel<<<(N_NODES + 127) / 128, 256, 0, stream>>>(h, W2, z);
    // out (= b2) += A @ z
    spmm_kernel<N_CLASS><<<(N_EDGES + 511) / 512, 64, 0, stream>>>(av, src, dst, z, out);
    // in-place log_softmax
    logsoftmax_kernel<<<(N_NODES + 255) / 256, 256, 0, stream>>>(out);
}